// SEIRNet_53996328845417
// MI455X (gfx1250) — compile-verified
//
#include <hip/hip_runtime.h>
#include <hip/hip_bf16.h>
#include <math.h>

// ---------------------------------------------------------------------------
// SEIR scan, split into:
//   Phase 1 (parallel, WMMA): b[t] = relu(W_b . x_t)^p  for all t  (x[5]:=0)
//   Phase 2 (serial wave32):  4-state nonlinear recurrence, coalesced I/O
// ---------------------------------------------------------------------------

typedef __attribute__((ext_vector_type(2))) float v2f;
typedef __attribute__((ext_vector_type(8))) float v8f;

// Single-instruction clamp to [0,1]: v_med3_f32
__device__ __forceinline__ float clamp01(float x) {
    return __builtin_amdgcn_fmed3f(x, 0.0f, 1.0f);
}

// ---------------------------------------------------------------------------
// Phase 1: per wave, one 16-row tile of X. Two accumulating
// V_WMMA_F32_16X16X4_F32 cover K=0..3 and K=4..5 (x[5] forced to 0).
// A 16x4 f32 layout (ISA 7.12.2): lanes 0-15 hold {K0,K1}, lanes 16-31 {K2,K3}.
// B 4x16 dual layout: VGPR0 = K0|K2 rows, VGPR1 = K1|K3 rows; all 16 columns
// carry the same W_b value, so every lane's D holds 8 valid dot results.
// ---------------------------------------------------------------------------
__global__ void seir_bexp_wmma_kernel(const float* __restrict__ X,
                                      const float* __restrict__ W_b,
                                      const float* __restrict__ p_exp,
                                      float* __restrict__ bbuf,
                                      long T)
{
    const int  lane = threadIdx.x & 31;
    const int  wave = threadIdx.x >> 5;
    const long base = ((long)blockIdx.x * (blockDim.x >> 5) + wave) * 16;
    if (base >= T) return;                       // wave-uniform: EXEC stays all-1s

    const bool hi = (lane >= 16);
    const int  m  = lane & 15;
    long row = base + m;
    if (row >= T) row = T - 1;                   // tail clamp (T%16==0 in practice)
    const float* xr = X + row * 6;

    v2f a1, a2;
    if (!hi) { a1.x = xr[0]; a1.y = xr[1]; a2.x = xr[4]; a2.y = 0.0f; }
    else     { a1.x = xr[2]; a1.y = xr[3]; a2.x = 0.0f;  a2.y = 0.0f; }

    const float w0 = W_b[0], w1 = W_b[1], w2 = W_b[2], w3 = W_b[3], w4 = W_b[4];
    v2f B1, B2;
    B1.x = hi ? w2 : w0;                         // K=0 | K=2 rows
    B1.y = hi ? w3 : w1;                         // K=1 | K=3 rows
    B2.x = hi ? 0.0f : w4;                       // second WMMA: K=4 row (K=6,7 zero)
    B2.y = 0.0f;                                 // x[5] is zeroed by the reference

    v8f c = {};
    c = __builtin_amdgcn_wmma_f32_16x16x4_f32(false, a1, false, B1, (short)0, c, false, false);
    c = __builtin_amdgcn_wmma_f32_16x16x4_f32(false, a2, false, B2, (short)0, c, false, false);

    const float pe = p_exp[0];
    float r[8];
#pragma unroll
    for (int i = 0; i < 8; ++i) {
        float y = fmaxf(c[i], 0.0f);
        // y^2.5 = y*y*sqrt(y)  (exact for the reference's p_exp); generic fallback
        r[i] = (pe == 2.5f) ? y * y * sqrtf(y)
                            : __builtin_exp2f(pe * __builtin_log2f(y));
    }

    // Lanes 0 and 16 hold M=0..7 / M=8..15 of column N=0: 16 contiguous results.
    if (m == 0) {
        long row0 = base + (hi ? 8 : 0);
        if (row0 + 8 <= T) {
            *(float4*)(bbuf + row0)     = make_float4(r[0], r[1], r[2], r[3]);
            *(float4*)(bbuf + row0 + 4) = make_float4(r[4], r[5], r[6], r[7]);
        }
    }
}

// ---------------------------------------------------------------------------
// Phase 2: one wave32 runs the serial recurrence. 32 b-values per coalesced
// load; v_readlane_b32 (const lane index) broadcasts each b off the critical
// path (no LDS traffic); all lanes compute the wave-uniform recurrence, lane j
// snapshots step j's state; then fully coalesced b128 + b32 stores.
// Critical path per step: v_med3-clamped FMA chains only.
// ---------------------------------------------------------------------------
__global__ void __launch_bounds__(32)
seir_scan_kernel(const float* __restrict__ bbuf,
                 const float* __restrict__ W_o,
                 const float* __restrict__ kP,  const float* __restrict__ sP,
                 const float* __restrict__ e0P, const float* __restrict__ i0P,
                 float* __restrict__ hiddens,   // [T*4]
                 float* __restrict__ outs,      // [T]
                 long T)
{
    const int lane = threadIdx.x & 31;
    const float k  = kP[0],  s  = sP[0];
    const float e0 = e0P[0], i0 = i0P[0];
    const float wo0 = W_o[0], wo1 = W_o[1], wo2 = W_o[2], wo3 = W_o[3];

    float h0 = i0;
    float h1 = 0.0f;
    float h2 = 1.0f - i0 - e0;
    float h3 = e0;

    for (long t0 = 0; t0 < T; t0 += 32) {
        long tl = t0 + lane;
        unsigned bvbits = __float_as_uint(bbuf[tl < T ? tl : (T - 1)]);
        if (t0 + 2048 < T)
            __builtin_prefetch(bbuf + t0 + 2048, 0, 1);   // global_prefetch_b8

        float sh0 = 0.f, sh1 = 0.f, sh2 = 0.f, sh3 = 0.f, so = 0.f;
#pragma unroll
        for (int j = 0; j < 32; ++j) {
            // v_readlane_b32 with immediate lane: uniform broadcast, no LDS
            float b  = __uint_as_float(__builtin_amdgcn_readlane((int)bvbits, j));
            float d1 = k * h0;
            float d3 = s * h3;
            float d2 = h0 * b * h2;
            float n0 = clamp01(h0 + d3 - d1);
            float n1 = clamp01(h1 + d1);
            float n2 = clamp01(h2 - d2);
            float n3 = clamp01(h3 + d2 - d3);
            h0 = n0; h1 = n1; h2 = n2; h3 = n3;
            if (lane == j) {                  // snapshot step j in lane j
                sh0 = n0; sh1 = n1; sh2 = n2; sh3 = n3;
                so  = wo0 * n0 + wo1 * n1 + wo2 * n2 + wo3 * n3;
            }
        }
        if (tl < T) {
            *(float4*)(hiddens + tl * 4) = make_float4(sh0, sh1, sh2, sh3);
            outs[tl] = so;
        }
    }
}

// ---------------------------------------------------------------------------
// Host side. Inputs (setup_inputs order):
//   0:X (T*6)  1:W_b(6)  2:W_o(4)  3:p_exp(1)  4:k  5:s  6:e0  7:i0
// Output: hiddens (T*4 floats) then outs (T floats), concatenated.
// ---------------------------------------------------------------------------
extern "C" void kernel_launch(void* const* d_in, const int* in_sizes, int n_in,
                              void* d_out, int out_size, void* d_ws, size_t ws_size,
                              hipStream_t stream)
{
    const float* X     = (const float*)d_in[0];
    const float* W_b   = (const float*)d_in[1];
    const float* W_o   = (const float*)d_in[2];
    const float* p_exp = (const float*)d_in[3];
    const float* kP    = (const float*)d_in[4];
    const float* sP    = (const float*)d_in[5];
    const float* e0P   = (const float*)d_in[6];
    const float* i0P   = (const float*)d_in[7];

    const long T = (long)in_sizes[0] / 6;

    float* hiddens = (float*)d_out;            // T*4
    float* outs    = hiddens + 4 * T;          // T
    // b staging: workspace if big enough, else the outs slice (read-before-write).
    float* bbuf = (ws_size >= (size_t)T * sizeof(float)) ? (float*)d_ws : outs;

    // Phase 1: 16 rows per wave, 8 waves per 256-thread block.
    const long waves  = (T + 15) / 16;
    const long blocks = (waves + 7) / 8;
    seir_bexp_wmma_kernel<<<dim3((unsigned)blocks), dim3(256), 0, stream>>>(
        X, W_b, p_exp, bbuf, T);

    // Phase 2: single wave, serial scan (stream-ordered after phase 1).
    seir_scan_kernel<<<dim3(1), dim3(32), 0, stream>>>(
        bbuf, W_o, kP, sP, e0P, i0P, hiddens, outs, T);
}